// Attention_2095944040585
// MI455X (gfx1250) — compile-verified
//
#include <hip/hip_runtime.h>
#include <math.h>

// Sizes for this problem instance
#define BBATCH 16
#define NSEQ   577
#define DIMM   768
#define HEADS  12
#define DH     64
#define INNER  768
#define QKVW   2304       // 3*INNER
#define TOPK   432        // int(577*0.75)
#define NTILE  37         // ceil(577/16)
#define KSTRIDE 68        // padded LDS row stride (dwords): 16 lanes -> 16 bank groups

typedef float v2f __attribute__((ext_vector_type(2)));
typedef float v8f __attribute__((ext_vector_type(8)));

__device__ __forceinline__ v8f wmma_f32(v2f a, v2f b, v8f c) {
  // V_WMMA_F32_16X16X4_F32 : D(16x16,f32) = A(16x4,f32) x B(4x16,f32) + C
  return __builtin_amdgcn_wmma_f32_16x16x4_f32(false, a, false, b, (short)0, c,
                                               false, false);
}

// Async copy of a 16x16-float tile (rows clamped to the last valid sequence
// row) from global memory into LDS using the gfx1250 async-DMA path.
// Each lane issues two GLOBAL_LOAD_ASYNC_TO_LDS_B128 (2 x 16B chunks);
// completion is tracked with ASYNCcnt (2 instructions per call).
__device__ __forceinline__ void async_tile(const float* __restrict__ gbase,
                                           int j0, float* dst, int lane) {
  #pragma unroll
  for (int p = 0; p < 2; ++p) {
    int chunk = lane + 32 * p;          // 64 chunks of 16B = 16 rows x 16 f32
    int row = chunk >> 2, c4 = chunk & 3;
    int jr = min(j0 + row, NSEQ - 1);
    const float* src = gbase + (size_t)jr * QKVW + c4 * 4;
    unsigned lds_addr = (unsigned)(size_t)(dst + row * KSTRIDE + c4 * 4);
    asm volatile("global_load_async_to_lds_b128 %0, %1, off"
                 :: "v"(lds_addr), "v"((unsigned long long)(size_t)src)
                 : "memory");
  }
}

// ---------------------------------------------------------------------------
// Kernel 1: QKV GEMM  Y[9232,2304] = X[9232,768] @ W[768,2304]
// One wave computes a 16x64 tile: 4 accumulators, 192 k-steps of 4.
// ---------------------------------------------------------------------------
__global__ __launch_bounds__(128) void qkv_gemm(const float* __restrict__ X,
                                                const float* __restrict__ W,
                                                float* __restrict__ Y) {
  int wid  = (blockIdx.x * blockDim.x + threadIdx.x) >> 5;   // 0..20771
  int lane = threadIdx.x & 31;
  int ln = lane & 15, hi = lane >> 4;
  int mt = wid / 36, nt = wid % 36;                          // 577 x 36 tiles
  int row = mt * 16 + ln;
  int n0  = nt * 64;
  const float* xr = X + (size_t)row * DIMM;
  v8f acc0 = {}, acc1 = {}, acc2 = {}, acc3 = {};
  for (int k = 0; k < DIMM / 4; ++k) {
    int kk = k * 4 + 2 * hi;
    v2f a = *(const v2f*)(xr + kk);                          // A: X[row][kk..kk+1]
    const float* w0 = W + (size_t)kk * QKVW + n0 + ln;
    const float* w1 = w0 + QKVW;
    v2f b;
    b.x = w0[0];  b.y = w1[0];  acc0 = wmma_f32(a, b, acc0);
    b.x = w0[16]; b.y = w1[16]; acc1 = wmma_f32(a, b, acc1);
    b.x = w0[32]; b.y = w1[32]; acc2 = wmma_f32(a, b, acc2);
    b.x = w0[48]; b.y = w1[48]; acc3 = wmma_f32(a, b, acc3);
  }
  int mbase = mt * 16 + 8 * hi;
  #pragma unroll
  for (int r = 0; r < 8; ++r) {
    size_t o = (size_t)(mbase + r) * QKVW + n0 + ln;
    Y[o]      = acc0[r];
    Y[o + 16] = acc1[r];
    Y[o + 32] = acc2[r];
    Y[o + 48] = acc3[r];
  }
}

// ---------------------------------------------------------------------------
// Kernel 2: per-row stats. One workgroup per (b,i), 12 waves = 12 heads.
// Computes head-mean se[j] (pre-diag-mask), exact 432nd-largest threshold via
// 32-bit radix select with wave32 ballots, and softmax (max m, denom l).
// ---------------------------------------------------------------------------
__global__ __launch_bounds__(384) void attn_stats(
    const float* __restrict__ QKV, const float* __restrict__ scale,
    float* __restrict__ SE, float* __restrict__ Tt, float* __restrict__ Mm,
    float* __restrict__ Ll) {
  __shared__ float qs[DIMM];
  __shared__ float sse[608];
  int i = blockIdx.x, b = blockIdx.y;
  int tid = threadIdx.x;
  int h = tid >> 5, lane = tid & 31;
  const float* qrow = QKV + ((size_t)b * NSEQ + i) * QKVW;   // q = first 768
  for (int idx = tid; idx < DIMM; idx += 384) qs[idx] = qrow[idx];
  for (int idx = tid; idx < 608; idx += 384) sse[idx] = 0.f;
  __syncthreads();

  float sc = scale[h];
  const float* kb = QKV + (size_t)b * NSEQ * QKVW + INNER + h * DH;
  float sv[19];
  unsigned um[19];
  #pragma unroll 1
  for (int t = 0; t < 19; ++t) {
    int j = lane + 32 * t;
    float s = -INFINITY;
    unsigned u = 0u;                 // excluded entries sort below everything
    if (j < NSEQ) {
      const float* kr = kb + (size_t)j * QKVW;
      float d = 0.f;
      #pragma unroll
      for (int dd = 0; dd < DH / 4; ++dd) {
        float4 kv = *(const float4*)(kr + dd * 4);
        float4 qv = *(const float4*)(qs + h * DH + dd * 4);
        d += kv.x * qv.x + kv.y * qv.y + kv.z * qv.z + kv.w * qv.w;
      }
      s = d * sc;
      atomicAdd(&sse[j], s);         // se uses PRE-mask scores (diag included)
      if (j != i) {                  // diagonal excluded from top-k
        unsigned bs = __float_as_uint(s);
        u = (bs & 0x80000000u) ? ~bs : (bs | 0x80000000u);   // order-preserving
      }
    }
    sv[t] = s;
    um[t] = u;
  }
  __syncthreads();

  // write se row (mean over 12 heads)
  float* serow = SE + ((size_t)b * NSEQ + i) * NSEQ;
  for (int idx = tid; idx < NSEQ; idx += 384)
    serow[idx] = sse[idx] * (1.f / 12.f);

  // radix-select: largest T with count(u >= T) >= TOPK  (== 432nd largest)
  unsigned T = 0u;
  for (int bit = 31; bit >= 0; --bit) {
    unsigned cand = T | (1u << bit);
    int c = 0;
    #pragma unroll
    for (int t = 0; t < 19; ++t) c += __popcll(__ballot(um[t] >= cand));
    if (c >= TOPK) T = cand;
  }
  float mx = -INFINITY;
  #pragma unroll
  for (int t = 0; t < 19; ++t)
    if (um[t] >= T)
      mx = fmaxf(mx, sv[t] + sse[lane + 32 * t] * (1.f / 12.f));
  for (int off = 16; off; off >>= 1) mx = fmaxf(mx, __shfl_xor(mx, off));
  float sm = 0.f;
  #pragma unroll
  for (int t = 0; t < 19; ++t)
    if (um[t] >= T)
      sm += __expf(sv[t] + sse[lane + 32 * t] * (1.f / 12.f) - mx);
  for (int off = 16; off; off >>= 1) sm += __shfl_xor(sm, off);
  if (lane == 0) {
    size_t o = ((size_t)b * HEADS + h) * NSEQ + i;
    unsigned tb = (T & 0x80000000u) ? (T ^ 0x80000000u) : ~T;  // back to f32
    Tt[o] = __uint_as_float(tb);
    Mm[o] = mx;
    Ll[o] = sm;
  }
}

// ---------------------------------------------------------------------------
// Kernel 3: flash-style apply. One wave per (b,h,16-row tile).
// K/V tiles are double-buffered in LDS via GLOBAL_LOAD_ASYNC_TO_LDS_B128:
// the next j-tile's copies are issued before computing the current one, and
// s_wait_asynccnt 4 keeps exactly one tile of copies in flight.
// S = Q.K^T via WMMA, threshold/exp in registers, P staged through LDS
// (C-layout -> A-layout), O += P.V via WMMA.
// ---------------------------------------------------------------------------
__global__ __launch_bounds__(32) void attn_av(
    const float* __restrict__ QKV, const float* __restrict__ scale,
    const float* __restrict__ SE, const float* __restrict__ Tt,
    const float* __restrict__ Mm, const float* __restrict__ Ll,
    float* __restrict__ Omid) {
  __shared__ __align__(16) float ldsK[2][16 * KSTRIDE];
  __shared__ __align__(16) float ldsV[2][16 * KSTRIDE];
  __shared__ float pls[16 * 17];
  int it = blockIdx.x, h = blockIdx.y, b = blockIdx.z;
  int lane = threadIdx.x, ln = lane & 15, hi = lane >> 4;
  int i0 = it * 16;
  float sc = scale[h];

  int qr = min(i0 + ln, NSEQ - 1);
  const float* qp = QKV + ((size_t)b * NSEQ + qr) * QKVW + h * DH;
  v2f qa[16];
  #pragma unroll
  for (int kk = 0; kk < 16; ++kk) qa[kk] = *(const v2f*)(qp + kk * 4 + 2 * hi);

  float tr[8], mr[8], li[8];
  int irow[8];
  const float* se_base = SE + (size_t)b * NSEQ * NSEQ;
  #pragma unroll
  for (int r = 0; r < 8; ++r) {
    int ir = min(i0 + r + 8 * hi, NSEQ - 1);
    irow[r] = ir;
    size_t o = ((size_t)b * HEADS + h) * NSEQ + ir;
    tr[r] = Tt[o]; mr[r] = Mm[o]; li[r] = 1.f / Ll[o];
  }

  v8f o0 = {}, o1 = {}, o2 = {}, o3 = {};
  const float* kbase = QKV + (size_t)b * NSEQ * QKVW + INNER + h * DH;
  const float* vbase = QKV + (size_t)b * NSEQ * QKVW + 2 * INNER + h * DH;

  // prologue: prefetch tile 0 into buffer 0 (4 async instructions)
  async_tile(kbase, 0, &ldsK[0][0], lane);
  async_tile(vbase, 0, &ldsV[0][0], lane);

  for (int jt = 0; jt < NTILE; ++jt) {
    int buf = jt & 1;
    if (jt + 1 < NTILE) {
      // prefetch next tile into the other buffer, then wait so that only the
      // 4 just-issued copies remain outstanding -> current tile is complete.
      async_tile(kbase, (jt + 1) * 16, &ldsK[buf ^ 1][0], lane);
      async_tile(vbase, (jt + 1) * 16, &ldsV[buf ^ 1][0], lane);
      asm volatile("s_wait_asynccnt 0x4" ::: "memory");
    } else {
      asm volatile("s_wait_asynccnt 0x0" ::: "memory");
    }

    int j0 = jt * 16;
    const float* Kb = &ldsK[buf][0];
    v8f s = {};
    #pragma unroll
    for (int kk = 0; kk < 16; ++kk) {
      v2f bk = *(const v2f*)(Kb + ln * KSTRIDE + kk * 4 + 2 * hi);
      s = wmma_f32(qa[kk], bk, s);
    }
    int j = j0 + ln;
    #pragma unroll
    for (int r = 0; r < 8; ++r) {
      int m = r + 8 * hi;
      int iRow = i0 + m;
      float sval = s[r] * sc;
      float se = se_base[(size_t)irow[r] * NSEQ + min(j, NSEQ - 1)];
      float p = 0.f;
      // diagonal re-masked (recomputed S is finite there); ragged edges -> 0
      if (j < NSEQ && iRow < NSEQ && j != iRow && sval >= tr[r])
        p = __expf(sval + se - mr[r]);
      pls[m * 17 + ln] = p;
    }
    __syncthreads();
    const float* Vb = &ldsV[buf][0];
    #pragma unroll
    for (int kk = 0; kk < 4; ++kk) {
      v2f pa;
      pa.x = pls[ln * 17 + kk * 4 + 2 * hi];
      pa.y = pls[ln * 17 + kk * 4 + 2 * hi + 1];
      const float* vr0 = Vb + (kk * 4 + 2 * hi) * KSTRIDE + ln;
      const float* vr1 = vr0 + KSTRIDE;
      v2f vb;
      vb.x = vr0[0];  vb.y = vr1[0];  o0 = wmma_f32(pa, vb, o0);
      vb.x = vr0[16]; vb.y = vr1[16]; o1 = wmma_f32(pa, vb, o1);
      vb.x = vr0[32]; vb.y = vr1[32]; o2 = wmma_f32(pa, vb, o2);
      vb.x = vr0[48]; vb.y = vr1[48]; o3 = wmma_f32(pa, vb, o3);
    }
    __syncthreads();
  }
  #pragma unroll
  for (int r = 0; r < 8; ++r) {
    int iRow = i0 + r + 8 * hi;
    if (iRow < NSEQ) {
      size_t o = ((size_t)b * NSEQ + iRow) * INNER + h * DH + ln;
      Omid[o]      = o0[r] * li[r];
      Omid[o + 16] = o1[r] * li[r];
      Omid[o + 32] = o2[r] * li[r];
      Omid[o + 48] = o3[r] * li[r];
    }
  }
}

// ---------------------------------------------------------------------------
// Kernel 4: output projection  Y[9232,768] = Omid[9232,768] @ Wout + bias
// ---------------------------------------------------------------------------
__global__ __launch_bounds__(128) void out_proj(const float* __restrict__ Xm,
                                                const float* __restrict__ W,
                                                const float* __restrict__ bias,
                                                float* __restrict__ Y) {
  int wid  = (blockIdx.x * blockDim.x + threadIdx.x) >> 5;   // 0..6923
  int lane = threadIdx.x & 31;
  int ln = lane & 15, hi = lane >> 4;
  int mt = wid / 12, nt = wid % 12;                          // 577 x 12 tiles
  int row = mt * 16 + ln;
  int n0  = nt * 64;
  const float* xr = Xm + (size_t)row * INNER;
  v8f acc0 = {}, acc1 = {}, acc2 = {}, acc3 = {};
  for (int k = 0; k < INNER / 4; ++k) {
    int kk = k * 4 + 2 * hi;
    v2f a = *(const v2f*)(xr + kk);
    const float* w0 = W + (size_t)kk * DIMM + n0 + ln;
    const float* w1 = w0 + DIMM;
    v2f bv;
    bv.x = w0[0];  bv.y = w1[0];  acc0 = wmma_f32(a, bv, acc0);
    bv.x = w0[16]; bv.y = w1[16]; acc1 = wmma_f32(a, bv, acc1);
    bv.x = w0[32]; bv.y = w1[32]; acc2 = wmma_f32(a, bv, acc2);
    bv.x = w0[48]; bv.y = w1[48]; acc3 = wmma_f32(a, bv, acc3);
  }
  int mbase = mt * 16 + 8 * hi;
  #pragma unroll
  for (int r = 0; r < 8; ++r) {
    size_t o = (size_t)(mbase + r) * DIMM + n0 + ln;
    Y[o]      = acc0[r] + bias[n0 + ln];
    Y[o + 16] = acc1[r] + bias[n0 + ln + 16];
    Y[o + 32] = acc2[r] + bias[n0 + ln + 32];
    Y[o + 48] = acc3[r] + bias[n0 + ln + 48];
  }
}

// ---------------------------------------------------------------------------
// Host launcher. Workspace layout (floats), total ~136 MB:
//   qkv  : B*N*2304          = 21,270,528
//   se   : B*N*N             =  5,326,864
//   t,m,l: 3 * B*H*N         =    332,352
//   omid : B*N*768           =  7,090,176
// ---------------------------------------------------------------------------
extern "C" void kernel_launch(void* const* d_in, const int* in_sizes, int n_in,
                              void* d_out, int out_size, void* d_ws,
                              size_t ws_size, hipStream_t stream) {
  const float* x     = (const float*)d_in[0];
  const float* wqkv  = (const float*)d_in[1];
  const float* scale = (const float*)d_in[2];
  const float* wout  = (const float*)d_in[3];
  const float* bout  = (const float*)d_in[4];
  float* out = (float*)d_out;
  float* ws  = (float*)d_ws;

  size_t off_qkv = 0;
  size_t off_se  = off_qkv + (size_t)BBATCH * NSEQ * QKVW;
  size_t off_t   = off_se  + (size_t)BBATCH * NSEQ * NSEQ;
  size_t off_m   = off_t   + (size_t)BBATCH * HEADS * NSEQ;
  size_t off_l   = off_m   + (size_t)BBATCH * HEADS * NSEQ;
  size_t off_mid = off_l   + (size_t)BBATCH * HEADS * NSEQ;

  // 1) QKV GEMM: 577*36 = 20772 waves / 4 waves per block
  qkv_gemm<<<5193, 128, 0, stream>>>(x, wqkv, ws + off_qkv);
  // 2) per-row stats (se, top-k threshold, softmax m & l)
  attn_stats<<<dim3(NSEQ, BBATCH), 384, 0, stream>>>(
      ws + off_qkv, scale, ws + off_se, ws + off_t, ws + off_m, ws + off_l);
  // 3) attention apply + attn@V (async double-buffered K/V in LDS)
  attn_av<<<dim3(NTILE, HEADS, BBATCH), 32, 0, stream>>>(
      ws + off_qkv, scale, ws + off_se, ws + off_t, ws + off_m, ws + off_l,
      ws + off_mid);
  // 4) output projection: 577*12 = 6924 waves / 4 per block
  out_proj<<<1731, 128, 0, stream>>>(ws + off_mid, wout, bout, out);
}